// CARAFE_Downsample_3_exp_42090679500896
// MI455X (gfx1250) — compile-verified
//
#include <hip/hip_runtime.h>
#include <hip/hip_bf16.h>

typedef __attribute__((ext_vector_type(2))) float v2f;
typedef __attribute__((ext_vector_type(8))) float v8f;

#define B_    8
#define CIN   256
#define CC    64
#define H_    128
#define W_    128
#define HW    (H_ * W_)
#define HP    64
#define WP    64
#define KK_   5
#define NOC   25   // K*K mask channels
#define NOCP  28   // padded to multiple of 4

// ---------------------------------------------------------------------------
// Kernel 1: 1x1 compression conv as WMMA f32 GEMM.
//   cx[b][h][w][cc] = sum_c x[b][c][h][w] * w_comp[cc][c] + b_comp[cc]
// M = b*HW+hw (16-row tiles per wave), K = CIN (steps of 4), N = CC (4 tiles).
// ---------------------------------------------------------------------------
__global__ __launch_bounds__(256) void comp_gemm_wmma(
    const float* __restrict__ x, const float* __restrict__ w,
    const float* __restrict__ bias, float* __restrict__ cx) {
  const int lane = threadIdx.x & 31;
  const int wave = threadIdx.x >> 5;
  const int ln   = lane & 15;   // M (A) / N (B,D) within tile
  const int hi   = lane >> 4;   // k-half select per ISA A/B layout
  const int m0   = blockIdx.x * 128 + wave * 16;   // 16-pixel row tile
  const int b    = m0 >> 14;                        // / HW (HW = 16384)
  const int hw   = m0 & (HW - 1);
  const int xbase = b * CIN * HW + hw + ln;         // x[b, 0, hw+ln]

  v8f acc[4] = {v8f{}, v8f{}, v8f{}, v8f{}};
  float bv[4];
#pragma unroll
  for (int t = 0; t < 4; ++t) bv[t] = bias[t * 16 + ln];

  for (int k0 = 0; k0 < CIN; k0 += 4) {
    const int kk = k0 + 2 * hi;                     // lanes 0-15: k0,k0+1; 16-31: k0+2,k0+3
    v2f a;
    a.x = x[xbase + kk * HW];
    a.y = x[xbase + (kk + 1) * HW];
#pragma unroll
    for (int t = 0; t < 4; ++t) {
      // B[k][n] = w_comp[n][k]; lanes mirror A with N = ln
      v2f bf = *(const v2f*)(w + (t * 16 + ln) * CIN + kk);
      acc[t] = __builtin_amdgcn_wmma_f32_16x16x4_f32(
          false, a, false, bf, (short)0, acc[t], false, false);
    }
  }
  // D layout: VGPR v, lanes 0-15 -> (M=v, N=ln); lanes 16-31 -> (M=v+8, N=ln)
#pragma unroll
  for (int t = 0; t < 4; ++t) {
#pragma unroll
    for (int v = 0; v < 8; ++v) {
      cx[(m0 + v + hi * 8) * CC + t * 16 + ln] = acc[t][v] + bv[t];
    }
  }
}

// ---------------------------------------------------------------------------
// Kernel 2: 3x3 stride-2 conv over cx (NHWC) + bias + scaled softmax over the
// 25 mask channels. Weights staged in LDS as [tap(9)][ic(64)][ocP(28)].
// One thread per (b, oh, ow); block = 64x4.
// ---------------------------------------------------------------------------
__global__ __launch_bounds__(256) void enc_softmax(
    const float* __restrict__ cx, const float* __restrict__ w_enc,
    const float* __restrict__ b_enc, const float* __restrict__ power_p,
    float* __restrict__ mask) {
  __shared__ float lw[9 * CC * NOCP];   // 64512 bytes

  const int tid = threadIdx.y * 64 + threadIdx.x;
  for (int idx = tid; idx < NOC * CC * 9; idx += 256) {
    const int oc = idx / (CC * 9);
    const int r  = idx - oc * CC * 9;
    const int ic = r / 9;
    const int p  = r - ic * 9;                      // dy*3+dx
    lw[(p * CC + ic) * NOCP + oc] = w_enc[idx];
  }
  for (int idx = tid; idx < 9 * CC * (NOCP - NOC); idx += 256) {
    const int p  = idx / (CC * 3);
    const int r  = idx - p * CC * 3;
    const int ic = r / 3;
    lw[(p * CC + ic) * NOCP + NOC + (r - ic * 3)] = 0.0f;
  }
  __syncthreads();

  const int ow = threadIdx.x;
  const int oh = blockIdx.x * 4 + threadIdx.y;
  const int b  = blockIdx.y;

  float acc[NOCP];
#pragma unroll
  for (int o = 0; o < NOCP; ++o) acc[o] = 0.0f;

  for (int dy = 0; dy < 3; ++dy) {
    const int ih = 2 * oh + dy - 1;
    if (ih < 0 || ih >= H_) continue;
    for (int dx = 0; dx < 3; ++dx) {
      const int iw = 2 * ow + dx - 1;
      if (iw < 0 || iw >= W_) continue;
      const float* cp = cx + ((b * H_ + ih) * W_ + iw) * CC;
      const float* wp = lw + (dy * 3 + dx) * CC * NOCP;
      for (int ic4 = 0; ic4 < CC / 4; ++ic4) {
        const float4 cv4 = *(const float4*)(cp + ic4 * 4);
#pragma unroll
        for (int e = 0; e < 4; ++e) {
          const float cv = (e == 0) ? cv4.x : (e == 1) ? cv4.y : (e == 2) ? cv4.z : cv4.w;
          const float* wr = wp + (ic4 * 4 + e) * NOCP;
#pragma unroll
          for (int o4 = 0; o4 < NOCP / 4; ++o4) {
            const float4 wv = *(const float4*)(wr + o4 * 4);
            acc[o4 * 4 + 0] += cv * wv.x;
            acc[o4 * 4 + 1] += cv * wv.y;
            acc[o4 * 4 + 2] += cv * wv.z;
            acc[o4 * 4 + 3] += cv * wv.w;
          }
        }
      }
    }
  }

  const float scale = expf(power_p[0]);
  float z[NOC];
  float mx = -3.4e38f;
#pragma unroll
  for (int o = 0; o < NOC; ++o) {
    z[o] = (acc[o] + b_enc[o]) * scale;
    mx = fmaxf(mx, z[o]);
  }
  float s = 0.0f;
#pragma unroll
  for (int o = 0; o < NOC; ++o) { z[o] = expf(z[o] - mx); s += z[o]; }
  const float inv = 1.0f / s;
#pragma unroll
  for (int o = 0; o < NOC; ++o) {
    mask[((b * NOC + o) * HP + oh) * WP + ow] = z[o] * inv;
  }
}

// ---------------------------------------------------------------------------
// Kernel 3: out[b][c][h][w] = sum_{i,j} xpad[b][c][2h+i-2][2w+j-2] *
//                             mask[b][i*5+j][h][w]
// Block = 64 threads along w; grid (HP, C, B). mask reads broadcast from L2.
// ---------------------------------------------------------------------------
__global__ __launch_bounds__(64) void recombine(
    const float* __restrict__ x, const float* __restrict__ mask,
    float* __restrict__ out) {
  const int w = threadIdx.x;
  const int h = blockIdx.x;
  const int c = blockIdx.y;
  const int b = blockIdx.z;
  const float* xp = x + (b * CIN + c) * HW;
  const float* mp = mask + (b * NOC * HP + h) * WP + w;

  float acc = 0.0f;
#pragma unroll
  for (int i = 0; i < KK_; ++i) {
    const int ih = 2 * h + i - 2;
    if (ih < 0 || ih >= H_) continue;
#pragma unroll
    for (int j = 0; j < KK_; ++j) {
      const int iw = 2 * w + j - 2;
      if (iw < 0 || iw >= W_) continue;
      acc += xp[ih * W_ + iw] * mp[(i * KK_ + j) * HP * WP];
    }
  }
  out[((b * CIN + c) * HP + h) * WP + w] = acc;
}

extern "C" void kernel_launch(void* const* d_in, const int* in_sizes, int n_in,
                              void* d_out, int out_size, void* d_ws, size_t ws_size,
                              hipStream_t stream) {
  const float* x       = (const float*)d_in[0];
  const float* w_comp  = (const float*)d_in[1];
  const float* b_comp  = (const float*)d_in[2];
  const float* w_enc   = (const float*)d_in[3];
  const float* b_enc   = (const float*)d_in[4];
  const float* power_p = (const float*)d_in[5];

  float* cx   = (float*)d_ws;                                   // 8*128*128*64 f32
  float* mask = (float*)((char*)d_ws + (size_t)B_ * HW * CC * 4); // 8*25*64*64 f32

  comp_gemm_wmma<<<dim3(1024), dim3(256), 0, stream>>>(x, w_comp, b_comp, cx);
  enc_softmax<<<dim3(16, 8), dim3(64, 4), 0, stream>>>(cx, w_enc, b_enc, power_p, mask);
  recombine<<<dim3(HP, CIN, B_), dim3(64), 0, stream>>>(x, mask, (float*)d_out);
}